// NodeModel_45990509805613
// MI455X (gfx1250) — compile-verified
//
#include <hip/hip_runtime.h>

typedef __attribute__((ext_vector_type(2))) float v2f;
typedef __attribute__((ext_vector_type(8))) float v8f;

#define NN 50000
#define NE 1600000
#define DN 64
#define DE 32
#define DG 32
#define HIDDEN 128
#define NF 128          // 64 + 32 + 32 concat features
#define ROWS 64         // nodes per block in MLP kernel
#define LDF 130         // padded LDS row stride (floats) to avoid bank conflicts

// ---------------- scatter-mean pipeline ----------------

__global__ void zero_ws(float* __restrict__ p, int n) {
  int i = blockIdx.x * blockDim.x + threadIdx.x;
  if (i < n) p[i] = 0.0f;
}

__global__ void scatter_edges(const float* __restrict__ edge_attr,
                              const int* __restrict__ edge_index,
                              float* __restrict__ sums,
                              float* __restrict__ counts) {
  long long t = (long long)blockIdx.x * blockDim.x + threadIdx.x;
  int e = (int)(t >> 5);
  if (e >= NE) return;
  int c = (int)(t & 31);
  int dst = edge_index[NE + e];  // row 1 of edge_index = dest
  atomicAdd(&sums[(long long)dst * DE + c], edge_attr[(long long)e * DE + c]);
  if (c == 0) atomicAdd(&counts[dst], 1.0f);
}

__global__ void seg_mean(const float* __restrict__ sums,
                         const float* __restrict__ counts,
                         float* __restrict__ eob) {
  int i = blockIdx.x * blockDim.x + threadIdx.x;
  if (i >= NN * DE) return;
  float cnt = counts[i >> 5];
  eob[i] = sums[i] / fmaxf(cnt, 1.0f);
}

// ---------------- fused concat + 2-layer MLP with f32 WMMA ----------------
// Per block: 4 wave32s, 64 node rows. Each wave computes one 16-row M-tile.
// A fragment (16x4 f32): lanes 0-15 hold {K, K+1} of row (lane&15),
//                        lanes 16-31 hold {K+2, K+3}.
// B fragment (4x16 f32): mirrored (row K / K+1 vs K+2 / K+3, col = lane&15).
// C/D (16x16 f32): VGPR v of lane -> D[v + 8*(lane>>4)][lane&15].

__global__ __launch_bounds__(128) void mlp_wmma(
    const float* __restrict__ x, const float* __restrict__ eob,
    const float* __restrict__ u, const int* __restrict__ batch,
    const float* __restrict__ W1, const float* __restrict__ b1,
    const float* __restrict__ W2, const float* __restrict__ b2,
    float* __restrict__ y) {
  __shared__ float feat[ROWS * LDF];
  __shared__ float hbuf[ROWS * LDF];

  const int tid = threadIdx.x;
  const int nb = blockIdx.x * ROWS;

  // Build the [ROWS x 128] concat feature tile in LDS: [x | edge_out_bar | u[batch]]
  for (int idx = tid; idx < ROWS * NF; idx += 128) {
    int r = idx >> 7;
    int c = idx & 127;
    int node = nb + r;
    float v = 0.0f;
    if (node < NN) {
      if (c < DN)           v = x[(long long)node * DN + c];
      else if (c < DN + DE) v = eob[(long long)node * DE + (c - DN)];
      else                  v = u[(long long)batch[node] * DG + (c - DN - DE)];
    }
    feat[r * LDF + c] = v;
  }
  __syncthreads();

  const int wave = tid >> 5;
  const int lane = tid & 31;
  const int half = lane >> 4;   // selects K+0..1 vs K+2..3 within a WMMA step
  const int nidx = lane & 15;   // N column within tile / M row for A fragment
  const int mrow0 = wave * 16;  // this wave's M tile

  // Layer 1: h = relu(feat @ W1 + b1), W1 is [128 x 128] row-major (k, n)
  for (int nt = 0; nt < HIDDEN / 16; ++nt) {
    const int ncol = nt * 16 + nidx;
    const float bias = b1[ncol];
    v8f acc;
#pragma unroll
    for (int v = 0; v < 8; ++v) acc[v] = bias;
#pragma unroll 4
    for (int ks = 0; ks < NF / 4; ++ks) {
      int kk = ks * 4 + 2 * half;
      v2f a = *(const v2f*)&feat[(mrow0 + nidx) * LDF + kk];
      v2f b;
      b.x = W1[kk * HIDDEN + ncol];
      b.y = W1[(kk + 1) * HIDDEN + ncol];
      acc = __builtin_amdgcn_wmma_f32_16x16x4_f32(false, a, false, b,
                                                  (short)0, acc, false, false);
    }
#pragma unroll
    for (int v = 0; v < 8; ++v) {
      float r = acc[v];
      hbuf[(mrow0 + v + 8 * half) * LDF + ncol] = r > 0.0f ? r : 0.0f;
    }
  }
  __syncthreads();

  // Layer 2: y = relu(h @ W2 + b2), W2 is [128 x 64] row-major (k, n)
  for (int nt = 0; nt < DN / 16; ++nt) {
    const int ncol = nt * 16 + nidx;
    const float bias = b2[ncol];
    v8f acc;
#pragma unroll
    for (int v = 0; v < 8; ++v) acc[v] = bias;
#pragma unroll 4
    for (int ks = 0; ks < HIDDEN / 4; ++ks) {
      int kk = ks * 4 + 2 * half;
      v2f a = *(const v2f*)&hbuf[(mrow0 + nidx) * LDF + kk];
      v2f b;
      b.x = W2[kk * DN + ncol];
      b.y = W2[(kk + 1) * DN + ncol];
      acc = __builtin_amdgcn_wmma_f32_16x16x4_f32(false, a, false, b,
                                                  (short)0, acc, false, false);
    }
#pragma unroll
    for (int v = 0; v < 8; ++v) {
      int node = nb + mrow0 + v + 8 * half;
      if (node < NN) {
        float r = acc[v];
        y[(long long)node * DN + ncol] = r > 0.0f ? r : 0.0f;
      }
    }
  }
}

// ---------------- launch ----------------

extern "C" void kernel_launch(void* const* d_in, const int* in_sizes, int n_in,
                              void* d_out, int out_size, void* d_ws, size_t ws_size,
                              hipStream_t stream) {
  const float* x     = (const float*)d_in[0];
  const int*   eidx  = (const int*)d_in[1];
  const float* eattr = (const float*)d_in[2];
  const float* u     = (const float*)d_in[3];
  const int*   batch = (const int*)d_in[4];
  const float* W1    = (const float*)d_in[5];
  const float* b1    = (const float*)d_in[6];
  const float* W2    = (const float*)d_in[7];
  const float* b2    = (const float*)d_in[8];

  float* y   = (float*)d_out;                    // output 0: [NN x 64]
  float* eob = (float*)d_out + (size_t)NN * DN;  // output 1: [NN x 32]

  float* sums   = (float*)d_ws;                  // [NN x 32]
  float* counts = sums + (size_t)NN * DE;        // [NN]

  int nz = NN * DE + NN;
  zero_ws<<<(nz + 255) / 256, 256, 0, stream>>>((float*)d_ws, nz);

  long long sthreads = (long long)NE * 32;
  scatter_edges<<<(int)((sthreads + 255) / 256), 256, 0, stream>>>(
      eattr, eidx, sums, counts);

  int mz = NN * DE;
  seg_mean<<<(mz + 255) / 256, 256, 0, stream>>>(sums, counts, eob);

  mlp_wmma<<<(NN + ROWS - 1) / ROWS, 128, 0, stream>>>(
      x, eob, u, batch, W1, b1, W2, b2, y);
}